// FlashDeepseekLayer_2585570312830
// MI455X (gfx1250) — compile-verified
//
#include <hip/hip_runtime.h>
#include <math.h>

#define HDIM 2048
#define IDIM 1408
#define NEXP 16
#define TOPK 2
#define SIDIM 2816
#define NTOK 1024
#define MAXSLOT 2304   /* 2048 assignments + up to 15 pad per expert, 16-aligned */
#define MAXTILE 144    /* MAXSLOT / 16 */

typedef __attribute__((ext_vector_type(2))) float v2f;
typedef __attribute__((ext_vector_type(8))) float v8f;

__device__ __forceinline__ v8f wmma_f32(v2f a, v2f b, v8f c) {
  // V_WMMA_F32_16X16X4_F32: D = A(16x4) * B(4x16) + C(16x16)
  return __builtin_amdgcn_wmma_f32_16x16x4_f32(false, a, false, b, (short)0, c,
                                               false, false);
}

__device__ __forceinline__ float silu_mul(float g, float u) {
  return g / (1.0f + __expf(-g)) * u;
}

// ---------------- Router: logits -> softmax -> top2 -> counts ----------------
__global__ void router_kernel(const float* __restrict__ x,
                              const float* __restrict__ gw,
                              int* __restrict__ counts,
                              int* __restrict__ topk_i,
                              float* __restrict__ topk_w) {
  const int t = blockIdx.x;
  const int lane = threadIdx.x;
  float logit = -1e30f;
  if (lane < NEXP) {
    const float* xr = x + (size_t)t * HDIM;
    const float* gr = gw + (size_t)lane * HDIM;
    float acc = 0.0f;
    for (int h = 0; h < HDIM; ++h) acc = fmaf(xr[h], gr[h], acc);
    logit = acc;
  }
  // softmax over lanes 0..15
  float m = logit;
  for (int off = 8; off; off >>= 1) m = fmaxf(m, __shfl_xor(m, off, 16));
  float p = (lane < NEXP) ? __expf(logit - m) : 0.0f;
  float s = p;
  for (int off = 8; off; off >>= 1) s += __shfl_xor(s, off, 16);
  float score = p / s;

  if (lane == 0) {
    float b1 = -1.0f, b2 = -1.0f;
    int i1 = 0, i2 = 0;
    for (int e = 0; e < NEXP; ++e) {
      float v = __shfl(score, e, 32);
      if (v > b1) { b2 = b1; i2 = i1; b1 = v; i1 = e; }
      else if (v > b2) { b2 = v; i2 = e; }
    }
    topk_i[t * 2 + 0] = i1; topk_w[t * 2 + 0] = b1;
    topk_i[t * 2 + 1] = i2; topk_w[t * 2 + 1] = b2;
    atomicAdd(&counts[i1], 1);
    atomicAdd(&counts[i2], 1);
  }
}

// ---------------- 16-aligned exclusive scan of expert counts ----------------
__global__ void scan_kernel(const int* __restrict__ counts, int* __restrict__ po) {
  if (threadIdx.x == 0 && blockIdx.x == 0) {
    int acc = 0;
    for (int e = 0; e < NEXP; ++e) {
      po[e] = acc;
      acc += ((counts[e] + 15) >> 4) << 4;
    }
    po[NEXP] = acc;
  }
}

// ---------------- Scatter tokens into padded per-expert slots ----------------
__global__ void build_lists_kernel(const int* __restrict__ topk_i,
                                   const float* __restrict__ topk_w,
                                   const int* __restrict__ po,
                                   int* __restrict__ cursor,
                                   int* __restrict__ tok_of_slot,
                                   float* __restrict__ w_of_slot) {
  const int id = blockIdx.x * blockDim.x + threadIdx.x;
  if (id >= NTOK * TOPK) return;
  const int t = id >> 1;
  const int e = topk_i[id];
  const int p = atomicAdd(&cursor[e], 1);
  const int slot = po[e] + p;
  tok_of_slot[slot] = t;
  w_of_slot[slot] = topk_w[id];
}

// -------- locate expert for a padded 16-row tile (po entries are 16-aligned) --------
__device__ __forceinline__ int find_expert(const int* __restrict__ po, int base) {
  for (int e = 0; e < NEXP; ++e)
    if (base >= po[e] && base < po[e + 1]) return e;
  return -1;
}

// ---------------- Expert gate-up GEMM + SwiGLU ----------------
// grid: (MAXTILE, IDIM/128), block 256 (8 waves). Wave covers 16 rows x 16 cols.
__global__ void expert_gu_kernel(const float* __restrict__ x,
                                 const float* __restrict__ wgu,
                                 const int* __restrict__ po,
                                 const int* __restrict__ counts,
                                 const int* __restrict__ tok_of_slot,
                                 float* __restrict__ hbuf) {
  const int tile = blockIdx.x;
  const int base = tile * 16;
  __shared__ int s_e, s_valid, s_tok[16];
  if (threadIdx.x == 0) {
    int e = find_expert(po, base);
    s_e = e;
    int v = 0;
    if (e >= 0) {
      v = counts[e] - (base - po[e]);
      v = v < 0 ? 0 : (v > 16 ? 16 : v);
    }
    s_valid = v;
  }
  __syncthreads();
  const int e = s_e;
  if (e < 0) return;
  const int valid = s_valid;
  if (threadIdx.x < 16)
    s_tok[threadIdx.x] = (threadIdx.x < valid) ? tok_of_slot[base + threadIdx.x] : 0;
  __syncthreads();

  const int lane = threadIdx.x & 31;
  const int wv = threadIdx.x >> 5;
  const int half = lane >> 4;
  const int col = lane & 15;
  const int n0 = blockIdx.y * 128 + wv * 16;  // column within [0, IDIM)

  const size_t ldb = 2 * IDIM;
  const float* Bg = wgu + (size_t)e * HDIM * ldb + n0 + col;
  const float* a_ptr = x + (size_t)s_tok[col] * HDIM + 2 * half;

  v8f accg = {};
  v8f accu = {};
  for (int k0 = 0; k0 < HDIM; k0 += 4) {
    v2f a = *(const v2f*)(a_ptr + k0);
    const float* bp = Bg + (size_t)(k0 + 2 * half) * ldb;
    __builtin_prefetch(bp + 16 * ldb, 0, 1);
    v2f bg = {bp[0], bp[ldb]};
    v2f bu = {bp[IDIM], bp[IDIM + ldb]};
    accg = wmma_f32(a, bg, accg);
    accu = wmma_f32(a, bu, accu);
  }
  for (int r = 0; r < 8; ++r) {
    int m = r + 8 * half;
    if (m < valid)
      hbuf[(size_t)(base + m) * IDIM + n0 + col] = silu_mul(accg[r], accu[r]);
  }
}

// ---------------- Expert down GEMM, scaled by combine weight, atomic into out ----------------
// grid: (MAXTILE, HDIM/128), block 256.
__global__ void expert_down_kernel(const float* __restrict__ hbuf,
                                   const float* __restrict__ wd,
                                   const int* __restrict__ po,
                                   const int* __restrict__ counts,
                                   const int* __restrict__ tok_of_slot,
                                   const float* __restrict__ w_of_slot,
                                   float* __restrict__ out) {
  const int tile = blockIdx.x;
  const int base = tile * 16;
  __shared__ int s_e, s_valid;
  if (threadIdx.x == 0) {
    int e = find_expert(po, base);
    s_e = e;
    int v = 0;
    if (e >= 0) {
      v = counts[e] - (base - po[e]);
      v = v < 0 ? 0 : (v > 16 ? 16 : v);
    }
    s_valid = v;
  }
  __syncthreads();
  const int e = s_e;
  if (e < 0) return;
  const int valid = s_valid;

  const int lane = threadIdx.x & 31;
  const int wv = threadIdx.x >> 5;
  const int half = lane >> 4;
  const int col = lane & 15;
  const int n0 = blockIdx.y * 128 + wv * 16;  // column within [0, HDIM)

  const float* B = wd + (size_t)e * IDIM * HDIM + n0 + col;
  const float* a_ptr = hbuf + (size_t)(base + col) * IDIM + 2 * half;

  v8f acc = {};
  for (int k0 = 0; k0 < IDIM; k0 += 4) {
    v2f a = *(const v2f*)(a_ptr + k0);
    const float* bp = B + (size_t)(k0 + 2 * half) * HDIM;
    __builtin_prefetch(bp + 16 * HDIM, 0, 1);
    v2f b = {bp[0], bp[HDIM]};
    acc = wmma_f32(a, b, acc);
  }
  for (int r = 0; r < 8; ++r) {
    int m = r + 8 * half;
    if (m < valid) {
      int tok = tok_of_slot[base + m];
      float w = w_of_slot[base + m];
      atomicAdd(&out[(size_t)tok * HDIM + n0 + col], acc[r] * w);
    }
  }
}

// ---------------- Shared expert gate-up + SwiGLU ----------------
// grid: (NTOK/16, SIDIM/128), block 256.
__global__ void shared_gu_kernel(const float* __restrict__ x,
                                 const float* __restrict__ wgu,
                                 float* __restrict__ hs) {
  const int lane = threadIdx.x & 31;
  const int wv = threadIdx.x >> 5;
  const int half = lane >> 4;
  const int col = lane & 15;
  const int m0 = blockIdx.x * 16;
  const int n0 = blockIdx.y * 128 + wv * 16;  // column within [0, SIDIM)

  const size_t ldb = 2 * SIDIM;
  const float* Bg = wgu + n0 + col;
  const float* a_ptr = x + (size_t)(m0 + col) * HDIM + 2 * half;

  v8f accg = {};
  v8f accu = {};
  for (int k0 = 0; k0 < HDIM; k0 += 4) {
    v2f a = *(const v2f*)(a_ptr + k0);
    const float* bp = Bg + (size_t)(k0 + 2 * half) * ldb;
    __builtin_prefetch(bp + 16 * ldb, 0, 1);
    v2f bg = {bp[0], bp[ldb]};
    v2f bu = {bp[SIDIM], bp[SIDIM + ldb]};
    accg = wmma_f32(a, bg, accg);
    accu = wmma_f32(a, bu, accu);
  }
  for (int r = 0; r < 8; ++r) {
    int m = r + 8 * half;
    hs[(size_t)(m0 + m) * SIDIM + n0 + col] = silu_mul(accg[r], accu[r]);
  }
}

// ---------------- Shared expert down GEMM, atomic into out ----------------
// grid: (NTOK/16, HDIM/128), block 256.
__global__ void shared_down_kernel(const float* __restrict__ hs,
                                   const float* __restrict__ wd,
                                   float* __restrict__ out) {
  const int lane = threadIdx.x & 31;
  const int wv = threadIdx.x >> 5;
  const int half = lane >> 4;
  const int col = lane & 15;
  const int m0 = blockIdx.x * 16;
  const int n0 = blockIdx.y * 128 + wv * 16;  // column within [0, HDIM)

  const float* B = wd + n0 + col;
  const float* a_ptr = hs + (size_t)(m0 + col) * SIDIM + 2 * half;

  v8f acc = {};
  for (int k0 = 0; k0 < SIDIM; k0 += 4) {
    v2f a = *(const v2f*)(a_ptr + k0);
    const float* bp = B + (size_t)(k0 + 2 * half) * HDIM;
    __builtin_prefetch(bp + 16 * HDIM, 0, 1);
    v2f b = {bp[0], bp[HDIM]};
    acc = wmma_f32(a, b, acc);
  }
  for (int r = 0; r < 8; ++r) {
    int m = r + 8 * half;
    atomicAdd(&out[(size_t)(m0 + m) * HDIM + n0 + col], acc[r]);
  }
}

extern "C" void kernel_launch(void* const* d_in, const int* in_sizes, int n_in,
                              void* d_out, int out_size, void* d_ws, size_t ws_size,
                              hipStream_t stream) {
  const float* x            = (const float*)d_in[0];  // (1,1024,2048)
  const float* gate_w       = (const float*)d_in[1];  // (16,2048)
  const float* w_gate_up    = (const float*)d_in[2];  // (16,2048,2816)
  const float* w_down       = (const float*)d_in[3];  // (16,1408,2048)
  const float* shared_gu_w  = (const float*)d_in[4];  // (2048,5632)
  const float* shared_dn_w  = (const float*)d_in[5];  // (2816,2048)
  float* out = (float*)d_out;

  char* ws = (char*)d_ws;
  int*   counts      = (int*)(ws + 0);          // 16 ints
  int*   cursor      = (int*)(ws + 64);         // 16 ints
  int*   po          = (int*)(ws + 128);        // 17 ints
  int*   topk_i      = (int*)(ws + 256);        // 2048 ints
  float* topk_w      = (float*)(ws + 256 + 8192);
  int*   tok_of_slot = (int*)(ws + 16640);      // MAXSLOT ints
  float* w_of_slot   = (float*)(ws + 16640 + MAXSLOT * 4);
  float* hbuf        = (float*)(ws + 35328);    // MAXSLOT * IDIM floats
  float* hs          = hbuf + (size_t)MAXSLOT * IDIM;  // NTOK * SIDIM floats

  hipMemsetAsync(ws, 0, 128, stream);  // counts + cursor
  hipMemsetAsync(out, 0, (size_t)NTOK * HDIM * sizeof(float), stream);

  router_kernel<<<NTOK, 32, 0, stream>>>(x, gate_w, counts, topk_i, topk_w);
  scan_kernel<<<1, 32, 0, stream>>>(counts, po);
  build_lists_kernel<<<(NTOK * TOPK + 255) / 256, 256, 0, stream>>>(
      topk_i, topk_w, po, cursor, tok_of_slot, w_of_slot);
  expert_gu_kernel<<<dim3(MAXTILE, IDIM / 128), 256, 0, stream>>>(
      x, w_gate_up, po, counts, tok_of_slot, hbuf);
  expert_down_kernel<<<dim3(MAXTILE, HDIM / 128), 256, 0, stream>>>(
      hbuf, w_down, po, counts, tok_of_slot, w_of_slot, out);
  shared_gu_kernel<<<dim3(NTOK / 16, SIDIM / 128), 256, 0, stream>>>(
      x, shared_gu_w, hs);
  shared_down_kernel<<<dim3(NTOK / 16, HDIM / 128), 256, 0, stream>>>(
      hs, shared_dn_w, out);
}